// GroupedQueryAttention_54915451847064
// MI455X (gfx1250) — compile-verified
//
#include <hip/hip_runtime.h>

// ---------------------------------------------------------------------------
// GroupedQueryAttention (B=2,S=2048,H=2048, 32 heads / 8 KV groups, head=64)
// fp32 end-to-end on V_WMMA_F32_16X16X4_F32, with double-buffered LDS staging
// via GLOBAL_LOAD_ASYNC_TO_LDS_B128 (ASYNCcnt) on CDNA5 / gfx1250.
// ---------------------------------------------------------------------------

typedef __attribute__((ext_vector_type(2))) float v2f;
typedef __attribute__((ext_vector_type(8))) float v8f;

#define HID    2048
#define SEQ    2048
#define NHEADS 32
#define HDIM   64
#define KVDIM  512   // GROUPS * HDIM

__device__ __forceinline__ v8f wmma_f32(v2f a, v2f b, v8f c) {
  // (neg_a, A, neg_b, B, c_mod, C, reuse_a, reuse_b)
  return __builtin_amdgcn_wmma_f32_16x16x4_f32(false, a, false, b, (short)0, c,
                                               false, false);
}

// LDS byte offset of a __shared__ object = low 32 bits of its flat address
// (LDS aperture maps addr[31:0] directly to the LDS offset).
__device__ __forceinline__ unsigned lds_off(const void* p) {
  return (unsigned)(unsigned long long)p;
}

// GVS-mode async global->LDS 16-byte copy, tracked by ASYNCcnt.
__device__ __forceinline__ void async_copy_b128(unsigned lds_dst,
                                                const float* sbase,
                                                unsigned byte_off) {
  asm volatile("global_load_async_to_lds_b128 %0, %1, %2"
               :
               : "v"(lds_dst), "v"(byte_off), "s"(sbase)
               : "memory");
}

#define WAIT_ASYNC_LE(n) asm volatile("s_wait_asynccnt " #n ::: "memory")

// ---------------------------------------------------------------------------
// Projection GEMM:  C[M,N] = A[M,K] @ W[K,N] + bias[N]
// 256 threads = 8 waves (2x4), block tile 64x128, wave tile 32x32.
// Double-buffered LDS, 3 async b128 per thread per 16-wide k-stage.
// ---------------------------------------------------------------------------
__global__ __launch_bounds__(256) void gemm_bias_f32(
    const float* __restrict__ A, const float* __restrict__ W,
    const float* __restrict__ bias, float* __restrict__ C,
    int M, int N, int K) {
  __shared__ __align__(16) float lA[2][64][20];   // 64x16, row pad 20 (80B)
  __shared__ __align__(16) float lB[2][16][128];  // 16x128

  const int tid  = threadIdx.x;
  const int lane = tid & 31;
  const int wave = tid >> 5;     // 0..7
  const int wm   = wave >> 2;    // 0..1 (M dir)
  const int wn   = wave & 3;     // 0..3 (N dir)
  const int hi   = lane >> 4;
  const int ln   = lane & 15;

  const int bm = blockIdx.y * 64;
  const int bn = blockIdx.x * 128;

  // cooperative tile-load coordinates
  const int ar = tid >> 2;        // 0..63
  const int ac = (tid & 3) * 4;   // 0,4,8,12
  const int br = tid >> 5;        // 0..7 (and +8)
  const int bc = (tid & 31) * 4;  // 0..124

  // per-thread LDS destinations for both buffers
  unsigned dA[2], dB0[2], dB1[2];
#pragma unroll
  for (int u = 0; u < 2; ++u) {
    dA[u]  = lds_off(&lA[u][ar][ac]);
    dB0[u] = lds_off(&lB[u][br][bc]);
    dB1[u] = lds_off(&lB[u][br + 8][bc]);
  }
  const float* Abase = A + (size_t)bm * K;  // uniform SGPR base
  const float* Wbase = W + bn;

  // prologue: stage 0 -> buffer 0
  async_copy_b128(dA[0], Abase, (unsigned)((ar * K + ac) * 4));
  async_copy_b128(dB0[0], Wbase, (unsigned)((br * N + bc) * 4));
  async_copy_b128(dB1[0], Wbase, (unsigned)(((br + 8) * N + bc) * 4));

  v8f acc[2][2] = {};
  int cur = 0;

  for (int k0 = 0; k0 < K; k0 += 16) {
    if (k0 + 16 < K) {  // issue next stage into the other buffer
      const int nxt = cur ^ 1;
      const int k1 = k0 + 16;
      async_copy_b128(dA[nxt], Abase, (unsigned)((ar * K + k1 + ac) * 4));
      async_copy_b128(dB0[nxt], Wbase, (unsigned)(((k1 + br) * N + bc) * 4));
      async_copy_b128(dB1[nxt], Wbase, (unsigned)(((k1 + br + 8) * N + bc) * 4));
      WAIT_ASYNC_LE(0x3);  // current stage's 3 copies complete
    } else {
      WAIT_ASYNC_LE(0x0);
    }
    __syncthreads();

    const float(&cA)[64][20] = lA[cur];
    const float(&cB)[16][128] = lB[cur];
#pragma unroll
    for (int kk = 0; kk < 4; ++kk) {
      const int kc = kk * 4 + hi * 2;
      v2f a0, a1, b0, b1;
      a0.x = cA[wm * 32 + ln][kc];      a0.y = cA[wm * 32 + ln][kc + 1];
      a1.x = cA[wm * 32 + 16 + ln][kc]; a1.y = cA[wm * 32 + 16 + ln][kc + 1];
      b0.x = cB[kc][wn * 32 + ln];      b0.y = cB[kc + 1][wn * 32 + ln];
      b1.x = cB[kc][wn * 32 + 16 + ln]; b1.y = cB[kc + 1][wn * 32 + 16 + ln];
      acc[0][0] = wmma_f32(a0, b0, acc[0][0]);
      acc[0][1] = wmma_f32(a0, b1, acc[0][1]);
      acc[1][0] = wmma_f32(a1, b0, acc[1][0]);
      acc[1][1] = wmma_f32(a1, b1, acc[1][1]);
    }
    __syncthreads();  // all waves done with buf[cur] before it is re-filled
    cur ^= 1;
  }

  const float bia0 = bias[bn + wn * 32 + ln];
  const float bia1 = bias[bn + wn * 32 + 16 + ln];
#pragma unroll
  for (int i = 0; i < 2; ++i) {
#pragma unroll
    for (int j = 0; j < 2; ++j) {
      const int col = bn + wn * 32 + j * 16 + ln;
      const float bj = j ? bia1 : bia0;
#pragma unroll
      for (int r = 0; r < 8; ++r) {
        const int row = bm + wm * 32 + i * 16 + r + 8 * hi;
        C[(size_t)row * N + col] = acc[i][j][r] + bj;
      }
    }
  }
}

// ---------------------------------------------------------------------------
// Attention:  out[b,q,h*64+d] = sum_k floor(q.k/8 | mask) * v
// 128 threads = 4 waves; each wave owns one 16-row query tile.
// K/V tiles double-buffered via async b128; 4 copies per thread per stage.
// ---------------------------------------------------------------------------
__global__ __launch_bounds__(128) void gqa_attn(
    const float* __restrict__ Q, const float* __restrict__ Kp,
    const float* __restrict__ V, const int* __restrict__ amask,
    float* __restrict__ out) {
  __shared__ __align__(16) float lK[2][16][76];  // 16x64, row pad 76 (304B)
  __shared__ __align__(16) float lV[2][16][76];
  __shared__ __align__(16) float lS[4][16][18];  // per-wave score scratch

  const int tid  = threadIdx.x;
  const int lane = tid & 31;
  const int w    = tid >> 5;  // wave 0..3
  const int hi   = lane >> 4;
  const int ln   = lane & 15;

  const int b = blockIdx.z;
  const int h = blockIdx.y;
  const int g = h >> 2;  // KV group (GROUP_SIZE = 4)
  const int q0 = (blockIdx.x * 4 + w) * 16;

  // Resident Q fragments: lane holds row ln, cols 4*kk + 2*hi + {0,1}
  const float* Qb = Q + ((size_t)(b * SEQ + q0)) * HID + h * HDIM;
  v2f qa[16];
#pragma unroll
  for (int kk = 0; kk < 16; ++kk) {
    const float* p = Qb + (size_t)ln * HID + kk * 4 + hi * 2;
    qa[kk].x = p[0];
    qa[kk].y = p[1];
  }

  const float* Kbase = Kp + ((size_t)b * SEQ) * KVDIM + g * HDIM;
  const float* Vbase = V + ((size_t)b * SEQ) * KVDIM + g * HDIM;

  // cooperative K/V tile-load coordinates
  const int r = tid >> 4;        // 0..7 (and +8)
  const int c = (tid & 15) * 4;  // 0..60

  unsigned dK0[2], dK1[2], dV0[2], dV1[2];
#pragma unroll
  for (int u = 0; u < 2; ++u) {
    dK0[u] = lds_off(&lK[u][r][c]);
    dK1[u] = lds_off(&lK[u][r + 8][c]);
    dV0[u] = lds_off(&lV[u][r][c]);
    dV1[u] = lds_off(&lV[u][r + 8][c]);
  }

  const int nkb = SEQ / 16;  // 128 key blocks

  // prologue: stage 0 -> buffer 0
  {
    const unsigned o0 = (unsigned)(((0 * 16 + r) * KVDIM + c) * 4);
    const unsigned o1 = (unsigned)(((0 * 16 + r + 8) * KVDIM + c) * 4);
    async_copy_b128(dK0[0], Kbase, o0);
    async_copy_b128(dK1[0], Kbase, o1);
    async_copy_b128(dV0[0], Vbase, o0);
    async_copy_b128(dV1[0], Vbase, o1);
  }

  v8f o[4] = {};
  int cur = 0;

  for (int kb = 0; kb < nkb; ++kb) {
    if (kb + 1 < nkb) {
      const int nxt = cur ^ 1;
      const unsigned o0 = (unsigned)((((kb + 1) * 16 + r) * KVDIM + c) * 4);
      const unsigned o1 = (unsigned)((((kb + 1) * 16 + r + 8) * KVDIM + c) * 4);
      async_copy_b128(dK0[nxt], Kbase, o0);
      async_copy_b128(dK1[nxt], Kbase, o1);
      async_copy_b128(dV0[nxt], Vbase, o0);
      async_copy_b128(dV1[nxt], Vbase, o1);
      WAIT_ASYNC_LE(0x4);  // current stage's 4 copies complete
    } else {
      WAIT_ASYNC_LE(0x0);
    }
    __syncthreads();

    const float(&cK)[16][76] = lK[cur];
    const float(&cV)[16][76] = lV[cur];

    // score tile S = Qtile(16x64) @ Ktile^T(64x16) : 16 WMMAs
    v8f sc = {};
#pragma unroll
    for (int kk = 0; kk < 16; ++kk) {
      const int kc = kk * 4 + hi * 2;
      v2f bf;
      bf.x = cK[ln][kc];  // B[k][n] = K[n][k]
      bf.y = cK[ln][kc + 1];
      sc = wmma_f32(qa[kk], bf, sc);
    }

    // floor(s/8) + mask  (all 8 accum rows share column n = ln)
    const int mv = amask[b * SEQ + kb * 16 + ln];
#pragma unroll
    for (int e = 0; e < 8; ++e) {
      const float fs = floorf(sc[e] * 0.125f);
      sc[e] = (mv != 0) ? fs : -__builtin_inff();
    }

    // C-layout -> A-layout via LDS round trip
#pragma unroll
    for (int e = 0; e < 8; ++e) lS[w][e + 8 * hi][ln] = sc[e];
    __syncthreads();

    // O(16x64) += S(16x16) @ Vtile(16x64) : 16 WMMAs
#pragma unroll
    for (int kk2 = 0; kk2 < 4; ++kk2) {
      const int kc = kk2 * 4 + hi * 2;
      v2f a;
      a.x = lS[w][ln][kc];
      a.y = lS[w][ln][kc + 1];
#pragma unroll
      for (int nt = 0; nt < 4; ++nt) {
        v2f bf;
        bf.x = cV[kc][nt * 16 + ln];
        bf.y = cV[kc + 1][nt * 16 + ln];
        o[nt] = wmma_f32(a, bf, o[nt]);
      }
    }
    __syncthreads();  // all waves done with buf[cur] before it is re-filled
    cur ^= 1;
  }

  // store O: row = q0 + e + 8*hi, col = h*64 + nt*16 + ln
  float* ob = out + ((size_t)(b * SEQ + q0)) * HID + h * HDIM;
#pragma unroll
  for (int nt = 0; nt < 4; ++nt) {
#pragma unroll
    for (int e = 0; e < 8; ++e) {
      ob[(size_t)(e + 8 * hi) * HID + nt * 16 + ln] = o[nt][e];
    }
  }
}

// ---------------------------------------------------------------------------
extern "C" void kernel_launch(void* const* d_in, const int* in_sizes, int n_in,
                              void* d_out, int out_size, void* d_ws,
                              size_t ws_size, hipStream_t stream) {
  const float* x   = (const float*)d_in[0];
  const int* amask = (const int*)d_in[1];
  const float* Wq  = (const float*)d_in[2];
  const float* bq  = (const float*)d_in[3];
  const float* Wk  = (const float*)d_in[4];
  const float* bk  = (const float*)d_in[5];
  const float* Wv  = (const float*)d_in[6];
  const float* bv  = (const float*)d_in[7];
  float* out = (float*)d_out;

  const int M = in_sizes[0] / HID;  // B * S = 4096
  const int B = M / SEQ;            // 2

  float* Qw = (float*)d_ws;            // M x 2048
  float* Kw = Qw + (size_t)M * HID;    // M x 512
  float* Vw = Kw + (size_t)M * KVDIM;  // M x 512

  gemm_bias_f32<<<dim3(HID / 128, M / 64), 256, 0, stream>>>(x, Wq, bq, Qw, M,
                                                             HID, HID);
  gemm_bias_f32<<<dim3(KVDIM / 128, M / 64), 256, 0, stream>>>(x, Wk, bk, Kw, M,
                                                               KVDIM, HID);
  gemm_bias_f32<<<dim3(KVDIM / 128, M / 64), 256, 0, stream>>>(x, Wv, bv, Vw, M,
                                                               KVDIM, HID);

  gqa_attn<<<dim3(SEQ / 64, NHEADS, B), 128, 0, stream>>>(Qw, Kw, Vw, amask,
                                                          out);
}